// GateSparsemaxK_979252543920
// MI455X (gfx1250) — compile-verified
//
#include <hip/hip_runtime.h>
#include <hip/hip_bf16.h>

// GateSparsemaxK for MI455X (gfx1250, wave32).
// Sort-free sparsemax: Newton on f(tau) = sum(relu(z - tau)) - 1, exact in a
// few iterations for piecewise-linear f. Memory-bound (384 MB -> ~16.5 us at
// 23.3 TB/s), so the CDNA5 feature exercised is async global->LDS staging
// (GLOBAL_LOAD_ASYNC_TO_LDS_B128 + s_wait_asynccnt) with double buffering.

#define R_DIM       4096
#define BLOCKT      256
#define VEC_STEPS   4                 // 4 x float4 per thread
#define EPT         16                // elements per thread = R_DIM / BLOCKT
#define RPB         4                 // rows per block (double-buffered)
#define NEWTON_MAX  32
#define INV_TAU     (1.0f / 0.7f)
#define ZNEG_SCALED (-1.0e9f / 0.7f)  // where(mask<=0.5) value after /tau

#if defined(__AMDGCN__) && __has_builtin(__builtin_amdgcn_global_load_async_to_lds_b128)
#define HAVE_ASYNC 1
#else
#define HAVE_ASYNC 0
#endif

typedef __attribute__((ext_vector_type(4))) float v4f;
typedef __attribute__((ext_vector_type(4))) int   v4i;
#if HAVE_ASYNC
typedef __attribute__((address_space(1))) v4i* gas_v4i_ptr;  // global (AS1)
typedef __attribute__((address_space(3))) v4i* lds_v4i_ptr;  // LDS    (AS3)
#endif

// Stage one 4096-float row into LDS: 256 threads x 4 x 16B, lane-contiguous
// (each instruction covers a 4 KB contiguous span). Each thread stages exactly
// the bytes it will later read, so only asynccnt (not a barrier) gates use.
__device__ __forceinline__ void stage_row(const float* __restrict__ g,
                                          float* l, int tid) {
#pragma unroll
  for (int j = 0; j < VEC_STEPS; ++j) {
    const int e = (j * BLOCKT + tid) * 4;
#if HAVE_ASYNC
    __builtin_amdgcn_global_load_async_to_lds_b128(
        (gas_v4i_ptr)(g + e), (lds_v4i_ptr)(l + e),
        /*imm offset*/ 0, /*cpol*/ 0);
#else
    *(v4f*)(l + e) = *(const v4f*)(g + e);
#endif
  }
}

template <int N>
__device__ __forceinline__ void wait_async() {
#if HAVE_ASYNC
#if __has_builtin(__builtin_amdgcn_s_wait_asynccnt)
  __builtin_amdgcn_s_wait_asynccnt(N);
#else
  asm volatile("s_wait_asynccnt %0" ::"i"(N) : "memory");
#endif
#endif
}

__global__ __launch_bounds__(BLOCKT) void gate_sparsemax_kernel(
    const float* __restrict__ s, const float* __restrict__ mask,
    const int* __restrict__ kptr, float* __restrict__ out, int nrows) {
  __shared__ float lds_s[2][R_DIM];   // 32 KB
  __shared__ float lds_m[2][R_DIM];   // 32 KB  (total 64 KB static LDS)

  const int tid = threadIdx.x;
  const float kf = (float)kptr[0];
  const long long base = (long long)blockIdx.x * RPB;
  if (base >= nrows) return;

  // Prefetch row 0 of this block's strip (8 async b128 per thread).
  stage_row(s + base * R_DIM, lds_s[0], tid);
  stage_row(mask + base * R_DIM, lds_m[0], tid);

  for (int r = 0; r < RPB; ++r) {
    const long long row = base + r;
    if (row >= nrows) break;
    const int cur = r & 1;

    // Issue prefetch of the next row into the other stage, then wait for the
    // current row's 8 (older, in-order) async loads.
    if ((r + 1 < RPB) && (row + 1 < nrows)) {
      stage_row(s + (row + 1) * R_DIM, lds_s[cur ^ 1], tid);
      stage_row(mask + (row + 1) * R_DIM, lds_m[cur ^ 1], tid);
      wait_async<8>();
    } else {
      wait_async<0>();
    }

    // z = where(mask > 0.5, s, BIG_NEG) / tau, kept in 16 VGPRs per thread.
    float z[EPT];
#pragma unroll
    for (int j = 0; j < VEC_STEPS; ++j) {
      const int e = (j * BLOCKT + tid) * 4;
      v4f sv = *(const v4f*)(&lds_s[cur][e]);
      v4f mv = *(const v4f*)(&lds_m[cur][e]);
#pragma unroll
      for (int q = 0; q < 4; ++q)
        z[j * 4 + q] = (mv[q] > 0.5f) ? (sv[q] * INV_TAU) : ZNEG_SCALED;
    }

    // This row's staged s-data is dead now; reuse its head as reduction
    // scratch (barrier below orders the reuse).
    float* red = lds_s[cur];
    __syncthreads();

    // Row max: in-thread -> wave32 xor-shuffle -> 8 wave partials in LDS.
    float mx = z[0];
#pragma unroll
    for (int j = 1; j < EPT; ++j) mx = fmaxf(mx, z[j]);
#pragma unroll
    for (int off = 16; off > 0; off >>= 1)
      mx = fmaxf(mx, __shfl_xor(mx, off, 32));
    if ((tid & 31) == 0) red[tid >> 5] = mx;
    __syncthreads();
    float rowmax = red[0];
#pragma unroll
    for (int w = 1; w < 8; ++w) rowmax = fmaxf(rowmax, red[w]);

    // Newton on sum(relu(z - tau)) = 1 from tau0 = max - 1 (monotone, exact
    // finite termination on piecewise-linear f; break test is block-uniform).
    float tau = rowmax - 1.0f;
    for (int it = 0; it < NEWTON_MAX; ++it) {
      float S = 0.0f, C = 0.0f;
#pragma unroll
      for (int j = 0; j < EPT; ++j)
        if (z[j] > tau) { S += z[j]; C += 1.0f; }
#pragma unroll
      for (int off = 16; off > 0; off >>= 1) {
        S += __shfl_xor(S, off, 32);
        C += __shfl_xor(C, off, 32);
      }
      __syncthreads();  // previous-round readers of red[] are done
      if ((tid & 31) == 0) {
        red[(tid >> 5) * 2 + 0] = S;
        red[(tid >> 5) * 2 + 1] = C;
      }
      __syncthreads();
      float St = 0.0f, Ct = 0.0f;
#pragma unroll
      for (int w = 0; w < 8; ++w) {
        St += red[w * 2 + 0];
        Ct += red[w * 2 + 1];
      }
      const float tn = (St - 1.0f) / Ct;  // Ct >= 1 always (max is in support)
      if (tn == tau) break;               // same on all threads: uniform branch
      tau = tn;
    }

    // p = min(k * relu(z - tau), 1), coalesced float4 stores.
    float* orow = out + row * (long long)R_DIM;
#pragma unroll
    for (int j = 0; j < VEC_STEPS; ++j) {
      const int e = (j * BLOCKT + tid) * 4;
      v4f o;
#pragma unroll
      for (int q = 0; q < 4; ++q)
        o[q] = fminf(kf * fmaxf(z[j * 4 + q] - tau, 0.0f), 1.0f);
      *(v4f*)(orow + e) = o;
    }

    // Next iteration's async staging reuses this iteration's scratch buffer:
    // make sure every thread is past its red[] reads first.
    __syncthreads();
  }
}

extern "C" void kernel_launch(void* const* d_in, const int* in_sizes, int n_in,
                              void* d_out, int out_size, void* d_ws,
                              size_t ws_size, hipStream_t stream) {
  const float* s = (const float*)d_in[0];
  const float* m = (const float*)d_in[1];
  const int* k = (const int*)d_in[2];
  float* out = (float*)d_out;

  const int nrows = in_sizes[0] / R_DIM;              // 8192
  const int grid = (nrows + RPB - 1) / RPB;           // 2048 blocks
  gate_sparsemax_kernel<<<grid, BLOCKT, 0, stream>>>(s, m, k, out, nrows);
}